// GNNEncoder_58600533786650
// MI455X (gfx1250) — compile-verified
//
#include <hip/hip_runtime.h>
#include <hip/hip_bf16.h>

#define HIDDEN 128
#define AW (HIDDEN + 4)   // padded LDS row stride (132 floats) -> conflict-free ds_load_b64

typedef __attribute__((ext_vector_type(2))) float v2f;
typedef __attribute__((ext_vector_type(8))) float v8f;

// ---------------------------------------------------------------------------
// Degree / normalization kernels
// ---------------------------------------------------------------------------
__global__ void k_init_deg(float* __restrict__ deg, int N) {
    int i = blockIdx.x * blockDim.x + threadIdx.x;
    if (i < N) deg[i] = 1.0f;  // self-loop contributes 1
}

__global__ void k_count_deg(const int* __restrict__ dst, float* __restrict__ deg, int E) {
    int e = blockIdx.x * blockDim.x + threadIdx.x;
    if (e < E) atomicAdd(&deg[dst[e]], 1.0f);
}

__global__ void k_deg_to_dinv(float* __restrict__ deg, int N) {
    int i = blockIdx.x * blockDim.x + threadIdx.x;
    if (i < N) deg[i] = rsqrtf(deg[i]);   // deg >= 1 always (self-loops)
}

__global__ void k_edge_norm(const int* __restrict__ src, const int* __restrict__ dst,
                            const float* __restrict__ dinv, float* __restrict__ norm, int E) {
    int e = blockIdx.x * blockDim.x + threadIdx.x;
    if (e < E) norm[e] = dinv[src[e]] * dinv[dst[e]];
}

// ---------------------------------------------------------------------------
// acc = bias (broadcast per feature), float4-vectorized
// ---------------------------------------------------------------------------
__global__ void k_init_acc(float4* __restrict__ acc, const float4* __restrict__ bias,
                           int total4) {
    int i = blockIdx.x * blockDim.x + threadIdx.x;
    if (i < total4) acc[i] = bias[i & (HIDDEN / 4 - 1)];
}

// ---------------------------------------------------------------------------
// h = X @ W via V_WMMA_F32_16X16X4_F32 (fp32 in/out).
// One block (8 waves) per 16-row strip of X. The A strip (16x128) and the
// whole W (128x128, stored TRANSPOSED) are staged in LDS once; each wave then
// computes one 16x16 output tile with a pure ds_load + wmma inner loop.
//
// ISA 7.12.2 layouts (wave32):
//   A 16x4 : lane L row = L&15 ; VGPR0 = K{0|2}, VGPR1 = K{1|3} by lane half
//   B 4x16 : lane L col = L&15 ; VGPR0 = K{0|2}, VGPR1 = K{1|3} by lane half
//   C/D    : VGPR r -> (M=r, N=lane) lanes 0-15, (M=r+8, N=lane-16)
// ---------------------------------------------------------------------------
__global__ __launch_bounds__(256) void k_gemm_wmma(const float* __restrict__ X,
                                                   const float* __restrict__ W,
                                                   float* __restrict__ Hout, int M) {
    __shared__ float sA[16 * AW];        //  8.4 KB: A strip, row-major, padded
    __shared__ float sWT[HIDDEN * AW];   // 67.6 KB: W transposed (sWT[col][k])

    const int tid    = threadIdx.x;
    const int tile_m = blockIdx.x;
    const int row0   = tile_m * 16;

    // ---- stage A strip: 16 rows x 128 cols = 512 float4, 2 per thread ----
    #pragma unroll
    for (int i = 0; i < 2; ++i) {
        int v  = tid + i * 256;          // 0..511
        int r  = v >> 5;                 // row within strip
        int c4 = v & 31;                 // float4 column
        int gr = row0 + r;
        float4 d = (gr < M) ? ((const float4*)(X + (size_t)gr * HIDDEN))[c4]
                            : make_float4(0.f, 0.f, 0.f, 0.f);
        float* p = &sA[r * AW + c4 * 4];
        p[0] = d.x; p[1] = d.y; p[2] = d.z; p[3] = d.w;
    }

    // ---- stage W transposed: 4096 float4, 16 per thread ----
    #pragma unroll
    for (int i = 0; i < 16; ++i) {
        int v  = tid + i * 256;          // 0..4095
        int r  = v >> 5;                 // W row (k index)
        int c4 = v & 31;                 // float4 column group
        float4 d = ((const float4*)W)[v];
        sWT[(c4 * 4 + 0) * AW + r] = d.x;
        sWT[(c4 * 4 + 1) * AW + r] = d.y;
        sWT[(c4 * 4 + 2) * AW + r] = d.z;
        sWT[(c4 * 4 + 3) * AW + r] = d.w;
    }
    __syncthreads();

    const int wave = tid >> 5;           // tile_n = wave (0..7)
    const int lane = tid & 31;
    const int half = lane >> 4;          // 0|1
    const int l15  = lane & 15;
    const int col  = wave * 16 + l15;    // B/D column
    const int kb   = half * 2;           // K sub-offset {0,2}

    v8f c = {};
    #pragma unroll
    for (int k = 0; k < HIDDEN; k += 4) {
        v2f a, b;
        const float* ap = &sA[l15 * AW + k + kb];     // ds_load_b64
        a.x = ap[0]; a.y = ap[1];
        const float* bp = &sWT[col * AW + k + kb];    // ds_load_b64 (transposed W)
        b.x = bp[0]; b.y = bp[1];
        c = __builtin_amdgcn_wmma_f32_16x16x4_f32(
                /*neg_a=*/false, a, /*neg_b=*/false, b,
                /*c_mod=*/(short)0, c, /*reuse_a=*/false, /*reuse_b=*/false);
    }

    const int out_row0 = row0 + half * 8;
    float* hp = Hout + (size_t)out_row0 * HIDDEN + col;
    #pragma unroll
    for (int r = 0; r < 8; ++r) {
        if (out_row0 + r < M) hp[(size_t)r * HIDDEN] = c[r];
    }
}

// ---------------------------------------------------------------------------
// Scatter: acc[dst, f] += h[src, f] * norm[e]
// One thread per (edge, float4-group): float4 gather + 4 global atomic f32 adds
// ---------------------------------------------------------------------------
__global__ void k_edge_scatter(const int* __restrict__ src, const int* __restrict__ dst,
                               const float* __restrict__ norm, const float* __restrict__ h,
                               float* __restrict__ acc, long long total4) {
    long long idx = (long long)blockIdx.x * blockDim.x + threadIdx.x;
    if (idx >= total4) return;
    int e  = (int)(idx >> 5);            // 32 float4 groups per edge
    int f4 = (int)(idx & 31);
    int s = src[e], d = dst[e];
    float w = norm[e];
    float4 v = ((const float4*)(h + (size_t)s * HIDDEN))[f4];
    float* ap = acc + (size_t)d * HIDDEN + f4 * 4;
    atomicAdd(ap + 0, v.x * w);
    atomicAdd(ap + 1, v.y * w);
    atomicAdd(ap + 2, v.z * w);
    atomicAdd(ap + 3, v.w * w);
}

// ---------------------------------------------------------------------------
// xout = relu(acc + h * dinv^2)  (self-loop folded in; runs after scatter on
// the same stream, so no race). float4-vectorized.
// ---------------------------------------------------------------------------
__global__ void k_combine(const float4* __restrict__ acc, const float4* __restrict__ h,
                          const float* __restrict__ dinv, float4* __restrict__ xout,
                          long long total4) {
    long long idx = (long long)blockIdx.x * blockDim.x + threadIdx.x;
    if (idx >= total4) return;
    int i = (int)(idx >> 5);             // node index
    float di = dinv[i];
    float w = di * di;
    float4 a = acc[idx], hh = h[idx];
    float4 r;
    r.x = fmaf(hh.x, w, a.x); r.x = r.x > 0.f ? r.x : 0.f;
    r.y = fmaf(hh.y, w, a.y); r.y = r.y > 0.f ? r.y : 0.f;
    r.z = fmaf(hh.z, w, a.z); r.z = r.z > 0.f ? r.z : 0.f;
    r.w = fmaf(hh.w, w, a.w); r.w = r.w > 0.f ? r.w : 0.f;
    xout[idx] = r;
}

// ---------------------------------------------------------------------------
static inline size_t align256(size_t x) { return (x + 255) & ~(size_t)255; }

extern "C" void kernel_launch(void* const* d_in, const int* in_sizes, int n_in,
                              void* d_out, int out_size, void* d_ws, size_t ws_size,
                              hipStream_t stream) {
    const int* edge_index  = (const int*)d_in[0];    // (2, E) flat
    const float* embedding = (const float*)d_in[1];  // (N, 128)
    const float* Ws        = (const float*)d_in[2];  // (3, 128, 128)
    const float* bs        = (const float*)d_in[3];  // (3, 128)

    const int E = in_sizes[0] / 2;
    const int N = in_sizes[1] / HIDDEN;
    const int L = in_sizes[3] / HIDDEN;              // 3 layers

    const int* src = edge_index;
    const int* dst = edge_index + E;

    // Workspace layout
    char* ws = (char*)d_ws;
    size_t off = 0;
    float* dinv = (float*)(ws + off); off += align256((size_t)N * sizeof(float));
    float* norm = (float*)(ws + off); off += align256((size_t)E * sizeof(float));
    float* hbuf = (float*)(ws + off); off += align256((size_t)N * HIDDEN * sizeof(float));
    float* xbuf = (float*)(ws + off); off += align256((size_t)N * HIDDEN * sizeof(float));
    float* acc  = (float*)d_out;                     // accumulator every layer

    const int T = 256;

    // --- degree + symmetric normalization ---
    k_init_deg   <<<(N + T - 1) / T, T, 0, stream>>>(dinv, N);
    k_count_deg  <<<(E + T - 1) / T, T, 0, stream>>>(dst, dinv, E);
    k_deg_to_dinv<<<(N + T - 1) / T, T, 0, stream>>>(dinv, N);
    k_edge_norm  <<<(E + T - 1) / T, T, 0, stream>>>(src, dst, dinv, norm, E);

    const long long nh4 = (long long)N * (HIDDEN / 4);   // node float4 elems
    const long long et4 = (long long)E * (HIDDEN / 4);   // edge float4 work items
    const int nh_blocks = (int)((nh4 + T - 1) / T);
    const int et_blocks = (int)((et4 + T - 1) / T);
    const int gemm_blocks = (N + 15) / 16;               // one block per 16-row strip

    for (int l = 0; l < L; ++l) {
        const float* xin  = (l == 0) ? embedding : xbuf;
        float*       xout = (l == L - 1) ? (float*)d_out : xbuf;
        const float* Wl = Ws + (size_t)l * HIDDEN * HIDDEN;
        const float* bl = bs + (size_t)l * HIDDEN;

        k_init_acc    <<<nh_blocks, T, 0, stream>>>((float4*)acc, (const float4*)bl, (int)nh4);
        k_gemm_wmma   <<<gemm_blocks, T, 0, stream>>>(xin, Wl, hbuf, N);
        k_edge_scatter<<<et_blocks, T, 0, stream>>>(src, dst, norm, hbuf, acc, et4);
        k_combine     <<<nh_blocks, T, 0, stream>>>((const float4*)acc, (const float4*)hbuf,
                                                    dinv, (float4*)xout, nh4);
    }
}